// BistableRecurrentCellLayer_53532472377581
// MI455X (gfx1250) — compile-verified
//
#include <hip/hip_runtime.h>
#include <hip/hip_bf16.h>

typedef float v2f __attribute__((ext_vector_type(2)));
typedef float v8f __attribute__((ext_vector_type(8)));

#define Bc 128
#define Tc 512
#define Dc 256
#define Hc 512

// D = A(16x4 f32) * B(4x16 f32) + C(16x16 f32)  -> v_wmma_f32_16x16x4_f32
__device__ __forceinline__ v8f wmma4(v2f a, v2f b, v8f c) {
  return __builtin_amdgcn_wmma_f32_16x16x4_f32(
      /*neg_a=*/false, a, /*neg_b=*/false, b,
      /*c_mod=*/(short)0, c, /*reuse_a=*/false, /*reuse_b=*/false);
}

// ---------------------------------------------------------------------------
// Phase 1: pr = x @ Kr + br, pz = x @ Kz + bz, ph = x @ Kh
//   x  : [B, T, D]  flattened rows m = b*T + t  (M = 65536, K = 256)
//   K* : [D, H]
//   p* : stored [T, B, H]  (scan-friendly layout), streamed out with NT stores
// One wave computes a 16(M) x 32(N) tile for all three outputs:
//   A fragment reused by 6 WMMAs per k-step (2 n-tiles x 3 matrices).
// x and K* stay regular-temporal so they remain L2-resident across the
// 16 blockIdx.y passes; projections bypass (NT) since they are read once,
// far later, by the scan kernel.
// ---------------------------------------------------------------------------
__global__ __launch_bounds__(128) void brc_proj_kernel(
    const float* __restrict__ x,
    const float* __restrict__ kr, const float* __restrict__ kz,
    const float* __restrict__ kh,
    const float* __restrict__ br, const float* __restrict__ bz,
    float* __restrict__ pr, float* __restrict__ pz, float* __restrict__ ph) {
  const int lane = threadIdx.x & 31;
  const int wave = threadIdx.x >> 5;
  const int l15  = lane & 15;
  const int hsel = lane >> 4;  // 0: lanes 0-15, 1: lanes 16-31

  const int m0 = (blockIdx.x * 4 + wave) * 16;
  const int n0 = blockIdx.y * 32;

  v8f cr0 = {}, cz0 = {}, ch0 = {};
  v8f cr1 = {}, cz1 = {}, ch1 = {};

  // A layout (16x4 f32): lane 0-15 -> K = k0,k0+1 ; lane 16-31 -> K = k0+2,k0+3
  const float* arow = x + (size_t)(m0 + l15) * Dc + hsel * 2;
  const int col0 = n0 + l15;
  const int col1 = col0 + 16;

  for (int k0 = 0; k0 < Dc; k0 += 4) {
    v2f a = *(const v2f*)(arow + k0);
    const int kb = k0 + hsel * 2;  // B layout mirrors A: rows kb, kb+1
    const size_t o0 = (size_t)kb * Hc + col0;
    const size_t o1 = (size_t)kb * Hc + col1;

    v2f b;
    b.x = kr[o0]; b.y = kr[o0 + Hc];  cr0 = wmma4(a, b, cr0);
    b.x = kz[o0]; b.y = kz[o0 + Hc];  cz0 = wmma4(a, b, cz0);
    b.x = kh[o0]; b.y = kh[o0 + Hc];  ch0 = wmma4(a, b, ch0);

    b.x = kr[o1]; b.y = kr[o1 + Hc];  cr1 = wmma4(a, b, cr1);
    b.x = kz[o1]; b.y = kz[o1 + Hc];  cz1 = wmma4(a, b, cz1);
    b.x = kh[o1]; b.y = kh[o1 + Hc];  ch1 = wmma4(a, b, ch1);
  }

  const float br0 = br[col0], br1 = br[col1];
  const float bz0 = bz[col0], bz1 = bz[col1];

  // C layout: VGPR i holds row (i + 8*hsel), col (lane&15)
#pragma unroll
  for (int i = 0; i < 8; ++i) {
    const int m  = m0 + i + hsel * 8;
    const int t  = m & (Tc - 1);  // m % T
    const int bb = m >> 9;        // m / T
    const size_t base = ((size_t)t * Bc + bb) * Hc;
    __builtin_nontemporal_store(cr0[i] + br0, pr + base + col0);
    __builtin_nontemporal_store(cz0[i] + bz0, pz + base + col0);
    __builtin_nontemporal_store(ch0[i],       ph + base + col0);
    __builtin_nontemporal_store(cr1[i] + br1, pr + base + col1);
    __builtin_nontemporal_store(cz1[i] + bz1, pz + base + col1);
    __builtin_nontemporal_store(ch1[i],       ph + base + col1);
  }
}

// ---------------------------------------------------------------------------
// Phase 2: scan. Recurrence is elementwise in (b,h) -> one thread per element,
// sequential over t. Loads/stores fully coalesced (stride B*H / H in t).
// All traffic is single-use streaming -> NT loads and NT stores.
// ---------------------------------------------------------------------------
__device__ __forceinline__ float fast_rcp(float x) {
  return __builtin_amdgcn_rcpf(x);
}
// overflow-safe tanh: 1 - 2/(exp(2x)+1); correct limits at +-inf
__device__ __forceinline__ float fast_tanh(float x) {
  return 1.0f - 2.0f * fast_rcp(__expf(2.0f * x) + 1.0f);
}
__device__ __forceinline__ float fast_sigmoid(float x) {
  return fast_rcp(1.0f + __expf(-x));
}

__global__ __launch_bounds__(256) void brc_scan_kernel(
    const float* __restrict__ pr, const float* __restrict__ pz,
    const float* __restrict__ ph,
    const float* __restrict__ h0, const float* __restrict__ mr,
    const float* __restrict__ mz, float* __restrict__ out) {
  const int idx = blockIdx.x * blockDim.x + threadIdx.x;  // over B*H
  const int n = idx & (Hc - 1);
  const int b = idx >> 9;  // idx / H

  float h = h0[idx];
  const float wr = mr[n];
  const float wz = mz[n];

  const size_t BH = (size_t)Bc * Hc;
  const float* prp = pr + idx;
  const float* pzp = pz + idx;
  const float* php = ph + idx;
  float* op = out + (size_t)b * Tc * Hc + n;

  for (int t = 0; t < Tc; ++t) {
    const float xr = __builtin_nontemporal_load(prp + (size_t)t * BH);
    const float xz = __builtin_nontemporal_load(pzp + (size_t)t * BH);
    const float xh = __builtin_nontemporal_load(php + (size_t)t * BH);

    const float r    = fast_tanh(xr + h * wr) + 1.0f;
    const float z    = fast_sigmoid(xz + h * wz);
    const float cand = fast_tanh(xh + r * h);
    h = z * h + (1.0f - z) * cand;

    __builtin_nontemporal_store(h, op + (size_t)t * Hc);
  }
}

extern "C" void kernel_launch(void* const* d_in, const int* in_sizes, int n_in,
                              void* d_out, int out_size, void* d_ws,
                              size_t ws_size, hipStream_t stream) {
  const float* x  = (const float*)d_in[0];
  const float* h0 = (const float*)d_in[1];
  const float* kr = (const float*)d_in[2];
  const float* kz = (const float*)d_in[3];
  const float* kh = (const float*)d_in[4];
  const float* mr = (const float*)d_in[5];
  const float* mz = (const float*)d_in[6];
  const float* br = (const float*)d_in[7];
  const float* bz = (const float*)d_in[8];
  float* out = (float*)d_out;

  const size_t plane = (size_t)Tc * Bc * Hc;  // elements per projection
  float* pr = (float*)d_ws;
  float* pz = pr + plane;
  float* ph = pz + plane;

  // GEMM: grid over (M/64 blocks of 4 waves x 16 rows, N/32)
  dim3 gProj((Bc * Tc) / 64, Hc / 32);  // (1024, 16)
  brc_proj_kernel<<<gProj, 128, 0, stream>>>(x, kr, kz, kh, br, bz, pr, pz, ph);

  // Scan: one thread per (b,h) element
  dim3 gScan((Bc * Hc) / 256);  // 256 blocks
  brc_scan_kernel<<<gScan, 256, 0, stream>>>(pr, pz, ph, h0, mr, mz, out);
}